// GaussianLoss_4071628996808
// MI455X (gfx1250) — compile-verified
//
#include <hip/hip_runtime.h>
#include <hip/hip_bf16.h>

typedef __attribute__((ext_vector_type(2))) float v2f;
typedef __attribute__((ext_vector_type(8))) float v8f;

#define BB 2
#define KK 64
#define NN 16
#define HH 128
#define WW 128
#define HWF 16384            // Hf*Wf
#define ROWS_PER_BLOCK 8
#define NBLK_Y (HH / ROWS_PER_BLOCK)   // 16
#define NBLK (BB * KK * NBLK_Y)        // 2048

#define LOG2E 1.4426950408889634f
#define LN2   0.6931471805599453f

// One block per (bk, 8-row chunk). One wave per row, 8 WMMA tiles of 16 pixels.
// WMMA computes D[m][j] = -dist^2(center m, pixel j) / (2 r^2) * log2(e),
// so g = sum_m exp2(D[m][j]) needs no per-element rescale before v_exp_f32.
__global__ __launch_bounds__(256) void gauss_bce_kernel(
    const float* __restrict__ centers, const float* __restrict__ radius,
    const float* __restrict__ mask,    const int*   __restrict__ ind,
    const float* __restrict__ target,  const float* __restrict__ peak,
    float* __restrict__ partial)
{
  const int bk   = blockIdx.x;          // 0..127  (b*64 + k)
  const int b    = bk >> 6;
  const int lane = threadIdx.x & 31;
  const int wave = threadIdx.x >> 5;    // 0..7
  const int m    = lane & 15;           // center index (A rows) / pixel column (B cols)

  const int   idx = ind[bk];
  const float r   = radius[b * HWF + idx];
  const float inv2 = LOG2E / (2.0f * r * r);   // fold log2(e) into the bilinear form
  const float pkx = peak[bk * 2 + 0];
  const float pky = peak[bk * 2 + 1];
  const float msk = mask[bk];

  // gathered center m for this (b,k): channels 2m, 2m+1
  const float px = centers[(b * 32 + 2 * m + 0) * HWF + idx] + pkx;
  const float py = centers[(b * 32 + 2 * m + 1) * HWF + idx] + pky;

  // A[m][0..3] = inv2 * { px, py, -(px^2+py^2), -1 }
  // lanes 0-15: VGPR0=K0, VGPR1=K1 ; lanes 16-31: VGPR0=K2, VGPR1=K3
  v2f A;
  if (lane < 16) { A.x = px * inv2;                   A.y = py * inv2; }
  else           { A.x = -(px * px + py * py) * inv2; A.y = -inv2;     }

  const int   y   = blockIdx.y * ROWS_PER_BLOCK + wave;
  const float yf  = (float)y;
  const float y2  = yf * yf;
  const float ty  = 2.0f * yf;
  const float* trow = target + (bk * HH + y) * WW;
  float acc = 0.0f;

  #pragma unroll
  for (int tx = 0; tx < WW / 16; ++tx) {
    const float xf = (float)(tx * 16 + m);
    // B[0..3][j] = { 2x_j, 2y, 1, x_j^2 + y^2 }
    v2f Bm;
    if (lane < 16) { Bm.x = 2.0f * xf; Bm.y = ty;            }
    else           { Bm.x = 1.0f;      Bm.y = xf * xf + y2;  }

    v8f c = {};
    c = __builtin_amdgcn_wmma_f32_16x16x4_f32(false, A, false, Bm,
                                              (short)0, c, false, false);
    float gp = 0.0f;
    #pragma unroll
    for (int q = 0; q < 8; ++q) gp += __builtin_amdgcn_exp2f(c[q]); // v_exp_f32, pre-scaled
    // column j split across lanes j and j+16 -> combine
    float g = gp + __shfl_xor(gp, 16, 32);

    if (lane < 16) {   // lanes 0-15 own the 16 pixel columns of this tile
      // sigmoid via raw v_exp_f32 + v_rcp_f32 (no IEEE divide sequence)
      float e   = __builtin_amdgcn_exp2f(-g * LOG2E);
      float pmv = __builtin_amdgcn_rcpf(1.0f + e);
      float p   = pmv * msk;
      float t   = trow[tx * 16 + m] * msk;
      float pc  = fminf(fmaxf(p, 1e-12f), 1.0f);
      float qc  = fminf(fmaxf(1.0f - p, 1e-12f), 1.0f);
      // -(t*ln pc + (1-t)*ln qc) = -LN2 * (t*log2 pc + (1-t)*log2 qc)
      float l2  = t * __builtin_amdgcn_logf(pc) + (1.0f - t) * __builtin_amdgcn_logf(qc);
      acc = fmaf(-LN2, l2, acc);
    }
  }

  __shared__ float sdata[256];
  sdata[threadIdx.x] = acc;
  __syncthreads();
  for (int s = 128; s > 0; s >>= 1) {
    if (threadIdx.x < s) sdata[threadIdx.x] += sdata[threadIdx.x + s];
    __syncthreads();
  }
  if (threadIdx.x == 0)
    partial[blockIdx.y * gridDim.x + blockIdx.x] = sdata[0];
}

__global__ __launch_bounds__(256) void gauss_reduce_kernel(
    const float* __restrict__ partial, float* __restrict__ out)
{
  __shared__ float sdata[256];
  float a = 0.0f;
  for (int i = threadIdx.x; i < NBLK; i += 256) a += partial[i];
  sdata[threadIdx.x] = a;
  __syncthreads();
  for (int s = 128; s > 0; s >>= 1) {
    if (threadIdx.x < s) sdata[threadIdx.x] += sdata[threadIdx.x + s];
    __syncthreads();
  }
  if (threadIdx.x == 0) {
    float loss = sdata[0] * (1.0f / (float)(BB * KK * HH * WW));
    out[0] = loss;
    out[1] = loss;
  }
}

extern "C" void kernel_launch(void* const* d_in, const int* in_sizes, int n_in,
                              void* d_out, int out_size, void* d_ws, size_t ws_size,
                              hipStream_t stream) {
  const float* centers = (const float*)d_in[0];
  const float* radius  = (const float*)d_in[1];
  const float* mask    = (const float*)d_in[2];
  const int*   ind     = (const int*)d_in[3];
  const float* target  = (const float*)d_in[4];
  const float* peak    = (const float*)d_in[5];
  float* partial = (float*)d_ws;   // NBLK floats, all written each call

  dim3 grid(BB * KK, NBLK_Y);
  gauss_bce_kernel<<<grid, 256, 0, stream>>>(centers, radius, mask, ind,
                                             target, peak, partial);
  gauss_reduce_kernel<<<1, 256, 0, stream>>>(partial, (float*)d_out);
}